// DGCNNABN_54606214201507
// MI455X (gfx1250) — compile-verified
//
#include <hip/hip_runtime.h>

typedef float v2f __attribute__((ext_vector_type(2)));
typedef float v8f __attribute__((ext_vector_type(8)));

#define BB   16
#define NN   1024
#define KNN  20
#define XCC  512
#define EMB  1024
#define SLOPE_C 0.2f
#define EPS_BN  1e-5f

__device__ __forceinline__ unsigned fkey(float x) {
    unsigned u = __float_as_uint(x);
    return (u & 0x80000000u) ? ~u : (u | 0x80000000u);
}
__device__ __forceinline__ float funkey(unsigned k) {
    unsigned u = (k & 0x80000000u) ? (k & 0x7fffffffu) : ~k;
    return __uint_as_float(u);
}
__device__ __forceinline__ float lrelu(float x) { return x > 0.f ? x : SLOPE_C * x; }

// ---------------------------------------------------------------------------
// Per-point squared norms: sq[b*N+n] = sum_c F[b,n,c]^2
// ---------------------------------------------------------------------------
__global__ void sq_kernel(const float* __restrict__ F, int stride, int off, int C,
                          float* __restrict__ sq) {
    int i = blockIdx.x * blockDim.x + threadIdx.x;
    if (i >= BB * NN) return;
    const float* row = F + (size_t)i * stride + off;
    float s = 0.f;
    for (int c = 0; c < C; ++c) { float v = row[c]; s += v * v; }
    sq[i] = s;
}

// ---------------------------------------------------------------------------
// Gram / negative squared distance via f32 WMMA:
//   pd[b,i,j] = 2*dot(F_i,F_j) - |F_i|^2 - |F_j|^2
// One wave per 16x16 tile. Main K-loop is unguarded (vector b64 loads);
// a single guarded tail step handles C % 4 != 0 (block 1, C=3, scalar rows).
// ---------------------------------------------------------------------------
__global__ void __launch_bounds__(256) gram_kernel(
    const float* __restrict__ F, int stride, int off, int C,
    const float* __restrict__ sq, float* __restrict__ pd) {
    int wave = threadIdx.x >> 5, lane = threadIdx.x & 31;
    int tile = blockIdx.x * 8 + wave;
    int jt = tile & 63;
    int it = (tile >> 6) & 63;
    int b  = tile >> 12;
    if (b >= BB) return;
    int lrow = lane & 15, lhi = lane >> 4;
    const float* Ai = F + ((size_t)(b * NN + it * 16 + lrow)) * stride + off; // A row
    const float* Bj = F + ((size_t)(b * NN + jt * 16 + lrow)) * stride + off; // B col
    v8f acc = {};
    int Cm = C & ~3;                 // unguarded K range
    int c = 0;
#pragma unroll 4
    for (; c < Cm; c += 4) {
        int k0 = c + 2 * lhi;
        v2f a  = *(const v2f*)&Ai[k0];   // 8B aligned whenever Cm > 0 (xc layout)
        v2f bf = *(const v2f*)&Bj[k0];
        acc = __builtin_amdgcn_wmma_f32_16x16x4_f32(false, a, false, bf,
                                                    (short)0, acc, false, false);
    }
    if (c < C) {                     // single guarded tail step
        int k0 = c + 2 * lhi;
        v2f a, bf;
        a.x  = (k0     < C) ? Ai[k0]     : 0.f;
        a.y  = (k0 + 1 < C) ? Ai[k0 + 1] : 0.f;
        bf.x = (k0     < C) ? Bj[k0]     : 0.f;
        bf.y = (k0 + 1 < C) ? Bj[k0 + 1] : 0.f;
        acc = __builtin_amdgcn_wmma_f32_16x16x4_f32(false, a, false, bf,
                                                    (short)0, acc, false, false);
    }
    float sqj = sq[b * NN + jt * 16 + lrow];
#pragma unroll
    for (int r = 0; r < 8; ++r) {
        int mi = it * 16 + r + 8 * lhi;
        float val = 2.f * acc[r] - sq[b * NN + mi] - sqj;
        pd[((size_t)b * NN + mi) * NN + jt * 16 + lrow] = val;
    }
}

// ---------------------------------------------------------------------------
// Top-20 per row (largest pd). One wave per row; 32 values/lane in registers,
// 20 iterations of wave-wide argmax with low-index tie-break (matches top_k).
// ---------------------------------------------------------------------------
__global__ void __launch_bounds__(256) topk_kernel(const float* __restrict__ pd,
                                                   int* __restrict__ idxb) {
    int wave = threadIdx.x >> 5, lane = threadIdx.x & 31;
    int row = blockIdx.x * 8 + wave;
    if (row >= BB * NN) return;
    const float* p = pd + (size_t)row * NN;
    float vals[32];
#pragma unroll
    for (int t = 0; t < 32; ++t) vals[t] = p[lane + 32 * t];
    for (int s = 0; s < KNN; ++s) {
        float lm = -__builtin_inff(); int li = 0x7fffffff;
#pragma unroll
        for (int t = 0; t < 32; ++t) {
            int j = lane + 32 * t;
            if (vals[t] > lm || (vals[t] == lm && j < li)) { lm = vals[t]; li = j; }
        }
        for (int o = 16; o >= 1; o >>= 1) {
            float ov = __shfl_xor(lm, o, 32);
            int   oi = __shfl_xor(li, o, 32);
            if (ov > lm || (ov == lm && oi < li)) { lm = ov; li = oi; }
        }
        if ((li & 31) == lane) vals[li >> 5] = -__builtin_inff();
        if (lane == 0) idxb[row * KNN + s] = li;
    }
}

// ---------------------------------------------------------------------------
// Init helpers
// ---------------------------------------------------------------------------
__global__ void init_stats_kernel(float* gsum, float* gsq, int O) {
    int i = blockIdx.x * blockDim.x + threadIdx.x;
    if (i < O) { gsum[i] = 0.f; gsq[i] = 0.f; }
}
__global__ void init_minmax_kernel(unsigned* ymaxk, unsigned* ymink, int n) {
    int i = blockIdx.x * blockDim.x + threadIdx.x;
    if (i < n) { ymaxk[i] = fkey(-__builtin_inff()); ymink[i] = fkey(__builtin_inff()); }
}

// ---------------------------------------------------------------------------
// Edge conv GEMM, fused gather + WMMA + k-reduction + BN-stat accumulation.
// One workgroup per (b,n): builds 32x(2*Cin) edge-feature tile in LDS
// (20 real edges + zero pad), 8 waves sweep 16x16 WMMA tiles over O.
// Main K-loop unguarded; single guarded tail (block 1 only, KD=6).
// ---------------------------------------------------------------------------
__global__ void __launch_bounds__(256) edge_gemm_kernel(
    const float* __restrict__ F, int stride, int off, int Cin, int O,
    const int* __restrict__ idxb, const float* __restrict__ W,
    float* __restrict__ gsum, float* __restrict__ gsq,
    unsigned* __restrict__ ymaxk, unsigned* __restrict__ ymink) {
    __shared__ float As[32 * 512];  // 64 KB: max KP = 2*128 = 256 fits
    __shared__ int snbr[KNN];
    int tid = threadIdx.x;
    int bn = blockIdx.x;            // b*N + n
    int b = bn >> 10;
    if (tid < KNN) snbr[tid] = idxb[bn * KNN + tid];
    __syncthreads();
    int KD = 2 * Cin;
    int KP = (KD + 3) & ~3;
    const float* cent = F + (size_t)bn * stride + off;
    for (int e = tid; e < 32 * KP; e += 256) {
        int m = e / KP, c = e % KP;
        float v = 0.f;
        if (m < KNN && c < KD) {
            const float* ng = F + ((size_t)(b * NN + snbr[m])) * stride + off;
            v = (c < Cin) ? (ng[c] - cent[c]) : cent[c - Cin];
        }
        As[e] = v;
    }
    __syncthreads();
    int wave = tid >> 5, lane = tid & 31;
    int lrow = lane & 15, lhi = lane >> 4;
    int ntN = O >> 4;
    int KDm = KD & ~3;               // unguarded K range
    for (int t = wave; t < 2 * ntN; t += 8) {
        int nt = t % ntN, mt = t / ntN;
        int o = nt * 16 + lrow;
        const float* Wo = W + (size_t)o * KD;
        const float* Am = &As[(mt * 16 + lrow) * KP];
        v8f acc = {};
        int c = 0;
#pragma unroll 4
        for (; c < KDm; c += 4) {
            int k0 = c + 2 * lhi;
            v2f a  = *(const v2f*)&Am[k0];
            v2f bf = *(const v2f*)&Wo[k0];   // 8B aligned whenever KDm > 0
            acc = __builtin_amdgcn_wmma_f32_16x16x4_f32(false, a, false, bf,
                                                        (short)0, acc, false, false);
        }
        if (c < KD) {                // guarded tail (KD = 6 case)
            int k0 = c + 2 * lhi;
            v2f a = *(const v2f*)&Am[k0];   // LDS is zero-padded to KP
            v2f bf;
            bf.x = (k0     < KD) ? Wo[k0]     : 0.f;
            bf.y = (k0 + 1 < KD) ? Wo[k0 + 1] : 0.f;
            acc = __builtin_amdgcn_wmma_f32_16x16x4_f32(false, a, false, bf,
                                                        (short)0, acc, false, false);
        }
        float lsum = 0.f, lsq = 0.f, lmax = -__builtin_inff(), lmin = __builtin_inff();
        int any = 0;
#pragma unroll
        for (int r = 0; r < 8; ++r) {
            int m = mt * 16 + r + 8 * lhi;   // edge index within this (b,n)
            if (m < KNN) {
                float v = acc[r];
                lsum += v; lsq += v * v;
                lmax = fmaxf(lmax, v); lmin = fminf(lmin, v);
                any = 1;
            }
        }
        lsum += __shfl_xor(lsum, 16, 32);
        lsq  += __shfl_xor(lsq, 16, 32);
        lmax = fmaxf(lmax, __shfl_xor(lmax, 16, 32));
        lmin = fminf(lmin, __shfl_xor(lmin, 16, 32));
        any |= __shfl_xor(any, 16, 32);
        if (lhi == 0 && any) {
            atomicAdd(&gsum[o], lsum);
            atomicAdd(&gsq[o],  lsq);
            atomicMax(&ymaxk[(size_t)bn * O + o], fkey(lmax));
            atomicMin(&ymink[(size_t)bn * O + o], fkey(lmin));
        }
    }
}

// ---------------------------------------------------------------------------
// BN + LeakyReLU + (max over k via sign of BN scale), write into concat slice.
// ---------------------------------------------------------------------------
__global__ void bn_finalize_kernel(
    const float* __restrict__ gsum, const float* __restrict__ gsq,
    const unsigned* __restrict__ ymaxk, const unsigned* __restrict__ ymink,
    const float* __restrict__ g, const float* __restrict__ beta,
    float* __restrict__ xc, int O, int chanOff, float invCnt) {
    int i = blockIdx.x * blockDim.x + threadIdx.x;
    if (i >= BB * NN * O) return;
    int o = i % O, bn = i / O;
    float mean = gsum[o] * invCnt;
    float var  = gsq[o] * invCnt - mean * mean;
    float s = g[o] * rsqrtf(var + EPS_BN);
    float t = beta[o] - mean * s;
    float v = (s >= 0.f) ? funkey(ymaxk[i]) : funkey(ymink[i]);
    xc[(size_t)bn * XCC + chanOff + o] = lrelu(s * v + t);
}

// ---------------------------------------------------------------------------
// Final 1x1 conv: y5 = xc(16384x512) * W5^T(512x1024), WMMA with LDS-staged A,
// fused BN-stat accumulation.
// ---------------------------------------------------------------------------
__global__ void __launch_bounds__(256) final_gemm_kernel(
    const float* __restrict__ xc, const float* __restrict__ W5,
    float* __restrict__ y5, float* __restrict__ gsum, float* __restrict__ gsq) {
    __shared__ float As[16 * XCC];  // 32 KB
    int tid = threadIdx.x;
    int m   = blockIdx.x >> 3;      // 16-row tile of (b*N+n)
    int ntg = blockIdx.x & 7;
    for (int e = tid; e < 16 * XCC; e += 256)
        As[e] = xc[((size_t)m * 16 + (e >> 9)) * XCC + (e & 511)];
    __syncthreads();
    int wave = tid >> 5, lane = tid & 31;
    int lrow = lane & 15, lhi = lane >> 4;
    int o = (ntg * 8 + wave) * 16 + lrow;
    const float* Wo = W5 + (size_t)o * XCC;
    v8f acc = {};
#pragma unroll 4
    for (int c = 0; c < XCC; c += 4) {
        int k0 = c + 2 * lhi;
        v2f a  = *(const v2f*)&As[lrow * XCC + k0];
        v2f bf = *(const v2f*)&Wo[k0];
        acc = __builtin_amdgcn_wmma_f32_16x16x4_f32(false, a, false, bf,
                                                    (short)0, acc, false, false);
    }
    float lsum = 0.f, lsq = 0.f;
#pragma unroll
    for (int r = 0; r < 8; ++r) {
        int row = m * 16 + r + 8 * lhi;
        y5[(size_t)row * EMB + o] = acc[r];
        lsum += acc[r]; lsq += acc[r] * acc[r];
    }
    lsum += __shfl_xor(lsum, 16, 32);
    lsq  += __shfl_xor(lsq, 16, 32);
    if (lhi == 0) { atomicAdd(&gsum[o], lsum); atomicAdd(&gsq[o], lsq); }
}

// ---------------------------------------------------------------------------
// Final BN + LeakyReLU + max/mean over N. One wave per (b, o).
// ---------------------------------------------------------------------------
__global__ void __launch_bounds__(256) final_reduce_kernel(
    const float* __restrict__ y5,
    const float* __restrict__ gsum, const float* __restrict__ gsq,
    const float* __restrict__ g5, const float* __restrict__ b5,
    float* __restrict__ out) {
    int wave = threadIdx.x >> 5, lane = threadIdx.x & 31;
    int id = blockIdx.x * 8 + wave;
    if (id >= BB * EMB) return;
    int b = id / EMB, o = id % EMB;
    const float invCnt = 1.f / (float)(BB * NN);
    float mean = gsum[o] * invCnt;
    float var  = gsq[o] * invCnt - mean * mean;
    float s = g5[o] * rsqrtf(var + EPS_BN);
    float t = b5[o] - mean * s;
    float mx = -__builtin_inff(), sm = 0.f;
    for (int n = lane; n < NN; n += 32) {
        float z = lrelu(s * y5[((size_t)b * NN + n) * EMB + o] + t);
        mx = fmaxf(mx, z); sm += z;
    }
    for (int off = 16; off >= 1; off >>= 1) {
        mx = fmaxf(mx, __shfl_xor(mx, off, 32));
        sm += __shfl_xor(sm, off, 32);
    }
    if (lane == 0) {
        out[(size_t)b * 2 * EMB + o]       = mx;
        out[(size_t)b * 2 * EMB + EMB + o] = sm * (1.f / NN);
    }
}

// ---------------------------------------------------------------------------
extern "C" void kernel_launch(void* const* d_in, const int* in_sizes, int n_in,
                              void* d_out, int out_size, void* d_ws, size_t ws_size,
                              hipStream_t stream) {
    (void)in_sizes; (void)n_in; (void)out_size; (void)ws_size;
    const float* x = (const float*)d_in[0];
    const float* W[5]  = {(const float*)d_in[1],  (const float*)d_in[4],
                          (const float*)d_in[7],  (const float*)d_in[10],
                          (const float*)d_in[13]};
    const float* G[5]  = {(const float*)d_in[2],  (const float*)d_in[5],
                          (const float*)d_in[8],  (const float*)d_in[11],
                          (const float*)d_in[14]};
    const float* Be[5] = {(const float*)d_in[3],  (const float*)d_in[6],
                          (const float*)d_in[9],  (const float*)d_in[12],
                          (const float*)d_in[15]};
    char* ws = (char*)d_ws;
    // Workspace layout (~131 MB):
    float*    xc    = (float*)(ws);                                  // 32 MB (B,N,512)
    float*    pd    = (float*)(ws + (size_t)32  * 1024 * 1024);      // 64 MB (B,N,N)
    float*    y5    = pd;                                            // reuse after last topk
    float*    sq    = (float*)(ws + (size_t)96  * 1024 * 1024);      // 64 KB
    int*      idxb  = (int*)  (ws + (size_t)97  * 1024 * 1024);      // 1.25 MB
    unsigned* ymaxk = (unsigned*)(ws + (size_t)99  * 1024 * 1024);   // 16 MB
    unsigned* ymink = (unsigned*)(ws + (size_t)115 * 1024 * 1024);   // 16 MB
    float*    gsum  = (float*)(ws + (size_t)131 * 1024 * 1024);      // 4 KB
    float*    gsq   = gsum + EMB;                                    // 4 KB

    struct Blk { const float* F; int stride, off, Cin, O, chanOff; };
    const Blk blks[4] = {
        { x,  3,   0,   3,   64,  0   },   // x  -> x1
        { xc, XCC, 0,   64,  64,  64  },   // x1 -> x2
        { xc, XCC, 64,  64,  128, 128 },   // x2 -> x3
        { xc, XCC, 128, 128, 256, 256 },   // x3 -> x4
    };
    const float invCntE = 1.f / (float)(BB * NN * KNN);

    for (int i = 0; i < 4; ++i) {
        const Blk& L = blks[i];
        sq_kernel<<<(BB * NN + 255) / 256, 256, 0, stream>>>(L.F, L.stride, L.off, L.Cin, sq);
        gram_kernel<<<BB * 64 * 64 / 8, 256, 0, stream>>>(L.F, L.stride, L.off, L.Cin, sq, pd);
        topk_kernel<<<BB * NN / 8, 256, 0, stream>>>(pd, idxb);
        init_stats_kernel<<<(L.O + 255) / 256, 256, 0, stream>>>(gsum, gsq, L.O);
        init_minmax_kernel<<<(BB * NN * L.O + 255) / 256, 256, 0, stream>>>(ymaxk, ymink, BB * NN * L.O);
        edge_gemm_kernel<<<BB * NN, 256, 0, stream>>>(L.F, L.stride, L.off, L.Cin, L.O,
                                                      idxb, W[i], gsum, gsq, ymaxk, ymink);
        bn_finalize_kernel<<<(BB * NN * L.O + 255) / 256, 256, 0, stream>>>(
            gsum, gsq, ymaxk, ymink, G[i], Be[i], xc, L.O, L.chanOff, invCntE);
    }
    init_stats_kernel<<<(EMB + 255) / 256, 256, 0, stream>>>(gsum, gsq, EMB);
    final_gemm_kernel<<<(BB * NN / 16) * 8, 256, 0, stream>>>(xc, W[4], y5, gsum, gsq);
    final_reduce_kernel<<<BB * EMB / 8, 256, 0, stream>>>(y5, gsum, gsq, G[4], Be[4],
                                                          (float*)d_out);
}